// ChamferLoss_15762529976904
// MI455X (gfx1250) — compile-verified
//
#include <hip/hip_runtime.h>
#include <math.h>

typedef __attribute__((ext_vector_type(2))) float v2f;
typedef __attribute__((ext_vector_type(8))) float v8f;

#define BATCH 2
#define NPTS  8192
#define CHUNK 2048          // pc2 points staged in LDS per pass (32 KB as float4)
#define TILES (CHUNK / 16)  // 128 column tiles per chunk
#define WAVES_PER_BLOCK 8   // 256 threads = 8 wave32

#define F_INF 0x7F800000u

// ---------------------------------------------------------------------------
// Kernel 0: init workspace (accumulator = 0, col-min array = +inf bits)
// ---------------------------------------------------------------------------
__global__ void chamfer_init(float* accum, unsigned* min2, int n) {
    int i = blockIdx.x * blockDim.x + threadIdx.x;
    if (i == 0) accum[0] = 0.0f;
    if (i < n) min2[i] = F_INF;
}

// ---------------------------------------------------------------------------
// Kernel 1: main chamfer. grid = (N/128, B), block = 256.
// Homogeneous K=4 trick: A row = (-2ax,-2ay,-2az, 1), B col = (bx,by,bz,|b|^2)
//   => A.B = -2(a.b) + |b|^2, and C[v] = |a_v|^2 is loop-invariant.
// WMMA therefore emits the full d2 tile with zero per-tile VALU prep beyond
// the two fragment-select cndmasks. Column mins reduce via LDS atomics,
// flushed to global once per chunk.
// ---------------------------------------------------------------------------
__global__ void __launch_bounds__(256)
chamfer_main(const float* __restrict__ pc1,
             const float* __restrict__ pc2,
             const float* __restrict__ flow,
             float* __restrict__ accum,
             unsigned* __restrict__ min2) {
    __shared__ float4   s_pts[CHUNK];    // {x, y, z, |p|^2} : 32 KB
    __shared__ unsigned s_colmin[CHUNK]; // ordered-uint float bits : 8 KB

    const int b    = blockIdx.y;
    const int lane = threadIdx.x & 31;
    const int wave = threadIdx.x >> 5;
    const bool hi  = lane >= 16;          // high half-wave holds K=2,3 of fragments
    const int lid  = lane & 15;

    // ----- Build A fragment: 16 rows of a = pc1+flow, scaled by -2, K3 = 1 -----
    const int row_base = (blockIdx.x * WAVES_PER_BLOCK + wave) * 16;
    const int m  = row_base + lid;
    const int gi = (b * NPTS + m) * 3;
    const float ax = pc1[gi + 0] + flow[gi + 0];
    const float ay = pc1[gi + 1] + flow[gi + 1];
    const float az = pc1[gi + 2] + flow[gi + 2];
    const float an = ax * ax + ay * ay + az * az;

    v2f afrag;
    afrag.x = -2.0f * (hi ? az : ax);   // K=0 (low half) / K=2 (high half)
    afrag.y = hi ? 1.0f : (-2.0f * ay); // K=1 (low half) / K=3 = 1 (homogeneous)

    // Loop-invariant C: c[v] = |a_row|^2, row M = v (lanes 0-15) / v+8 (16-31)
    v8f cbase;
#pragma unroll
    for (int v = 0; v < 8; ++v)
        cbase[v] = __shfl(an, v + ((lane & 16) >> 1), 32);

    float rowmin[8];
#pragma unroll
    for (int v = 0; v < 8; ++v) rowmin[v] = __builtin_inff();

    for (int chunk = 0; chunk < NPTS / CHUNK; ++chunk) {
        __syncthreads();
        // cooperative stage of pc2 chunk into LDS + reset column-min slots
        for (int i = threadIdx.x; i < CHUNK; i += blockDim.x) {
            int gp = (b * NPTS + chunk * CHUNK + i) * 3;
            float x = pc2[gp + 0], y = pc2[gp + 1], z = pc2[gp + 2];
            s_pts[i]    = make_float4(x, y, z, x * x + y * y + z * z);
            s_colmin[i] = F_INF;
        }
        __syncthreads();

        for (int t = 0; t < TILES; ++t) {
            const float4 p = s_pts[t * 16 + lid];  // ds_load_b128, conflict-free
            v2f bfrag;
            bfrag.x = hi ? p.z : p.x;  // K=0 / K=2
            bfrag.y = hi ? p.w : p.y;  // K=1 / K=3 = |b|^2 (homogeneous)

            // d2 tile straight out of the matrix unit:
            // d2[M][N] = -2*a_M.b_N + |b_N|^2 + |a_M|^2
            v8f d = __builtin_amdgcn_wmma_f32_16x16x4_f32(
                    /*neg_a=*/false, afrag, /*neg_b=*/false, bfrag,
                    /*c_mod=*/(short)0, cbase, /*reuse_a=*/false, /*reuse_b=*/false);

            float colmin = __builtin_inff();
#pragma unroll
            for (int v = 0; v < 8; ++v) {
                rowmin[v] = fminf(rowmin[v], d[v]);
                colmin    = fminf(colmin, d[v]);
            }
            // LDS atomic min (ordered-uint on nonneg floats). Lanes L and L+16
            // target the same slot (rows v and v+8); the LDS atomic unit
            // serializes the 2-way collision.
            colmin = fmaxf(colmin, 0.0f);
            atomicMin(&s_colmin[t * 16 + lid], __float_as_uint(colmin));
        }

        __syncthreads();
        // flush this chunk's column mins to global (1 atomic per column/block)
        for (int i = threadIdx.x; i < CHUNK; i += blockDim.x)
            atomicMin(&min2[b * NPTS + chunk * CHUNK + i], s_colmin[i]);
    }

    // ----- Row-direction: butterfly min across each 16-lane half, then sqrt -----
    float s = 0.0f;
#pragma unroll
    for (int v = 0; v < 8; ++v) {
        float r = rowmin[v];
        r = fminf(r, __shfl_xor(r, 8, 32));
        r = fminf(r, __shfl_xor(r, 4, 32));
        r = fminf(r, __shfl_xor(r, 2, 32));
        r = fminf(r, __shfl_xor(r, 1, 32));
        if (lane == 0 || lane == 16)  // lane0: row v, lane16: row v+8
            s += sqrtf(fmaxf(r, 0.0f));
    }
    if (lane == 0 || lane == 16) atomicAdd(accum, s);
}

// ---------------------------------------------------------------------------
// Kernel 2: finalize. sum sqrt(colmin) over all B*N, combine, mean.
// ---------------------------------------------------------------------------
__global__ void chamfer_finalize(const float* __restrict__ accum,
                                 const unsigned* __restrict__ min2,
                                 float* __restrict__ out) {
    __shared__ float red[256];
    float s = 0.0f;
    for (int i = threadIdx.x; i < BATCH * NPTS; i += 256)
        s += sqrtf(fmaxf(__uint_as_float(min2[i]), 0.0f));
    red[threadIdx.x] = s;
    __syncthreads();
    for (int off = 128; off > 0; off >>= 1) {
        if ((int)threadIdx.x < off) red[threadIdx.x] += red[threadIdx.x + off];
        __syncthreads();
    }
    if (threadIdx.x == 0)
        out[0] = (red[0] + accum[0]) / (float)(BATCH * NPTS);
}

// ---------------------------------------------------------------------------
extern "C" void kernel_launch(void* const* d_in, const int* in_sizes, int n_in,
                              void* d_out, int out_size, void* d_ws, size_t ws_size,
                              hipStream_t stream) {
    const float* pc1  = (const float*)d_in[0];
    const float* pc2  = (const float*)d_in[1];
    const float* flow = (const float*)d_in[2];
    float* out = (float*)d_out;

    // workspace layout: [0..255] accumulator (float at 0), then B*N uint col-mins
    float*    accum = (float*)d_ws;
    unsigned* min2  = (unsigned*)((char*)d_ws + 256);

    chamfer_init<<<(BATCH * NPTS + 255) / 256, 256, 0, stream>>>(
        accum, min2, BATCH * NPTS);

    dim3 grid(NPTS / (WAVES_PER_BLOCK * 16), BATCH);  // (64, 2)
    chamfer_main<<<grid, WAVES_PER_BLOCK * 32, 0, stream>>>(
        pc1, pc2, flow, accum, min2);

    chamfer_finalize<<<1, 256, 0, stream>>>(accum, min2, out);
}